// MSELoss2D_rand_2783138808370
// MI455X (gfx1250) — compile-verified
//
#include <hip/hip_runtime.h>

// MI455X / gfx1250, wave32. Bandwidth-bound weighted-MSE over the 3x3 patch
// expansion, collapsed to a single pass over pixels with closed-form
// multiplicity weights w(k) = min(k+1, 3, 1024-k).
//
// Data path: non-temporal global_load_b128 (read-once streaming, keep L2
// clean) -> v_pk_fma_f32 accumulate -> wave32 reduction on the fp32 WMMA pipe
// (V_WMMA_F32_16X16X4_F32, B = ones -> row sums) -> LDS across the block's
// 8 waves -> one device-scope global_atomic_add_f32 per block.

typedef __attribute__((ext_vector_type(2))) float v2f;
typedef __attribute__((ext_vector_type(4))) float v4f;
typedef __attribute__((ext_vector_type(8))) float v8f;

__device__ __forceinline__ float wgt(int i) {
    // number of 3-wide windows covering index i in a length-1024 axis
    int w = i + 1;
    if (w > 3) w = 3;
    int w2 = 1024 - i;
    if (w2 < w) w = w2;
    return (float)w;
}

__global__ void zero_accum_kernel(float* __restrict__ acc) {
    if (threadIdx.x == 0 && blockIdx.x == 0) acc[0] = 0.0f;
}

__global__ __launch_bounds__(256) void mse_patch_wsum_kernel(
        const float* __restrict__ xg,
        const float* __restrict__ yg,
        float* __restrict__ accum) {
    constexpr int kTotal4 = (4 * 1024 * 1024) / 4;   // float4 chunks
    const int nthreads = (int)(gridDim.x * blockDim.x);
    const int tid      = (int)(blockIdx.x * blockDim.x + threadIdx.x);

    const v4f* __restrict__ x4 = (const v4f*)xg;
    const v4f* __restrict__ y4 = (const v4f*)yg;

    // Two accumulators: ILP in the loop AND they map exactly onto the two
    // A-matrix VGPRs of the 16x4 f32 WMMA tile used for the wave reduction.
    float acc0 = 0.0f, acc1 = 0.0f;

    for (int q = tid; q < kTotal4; q += nthreads) {
        // Prime L2 for the next grid-stride iteration (speculative prefetch:
        // OOB translations are silently dropped).
        __builtin_prefetch(&x4[q + nthreads], 0, 0);
        __builtin_prefetch(&y4[q + nthreads], 0, 0);

        // Read-once streaming loads: non-temporal hint keeps WGP$/L2 clean.
        const v4f xv = __builtin_nontemporal_load(&x4[q]);  // global_load_b128 th:NT
        const v4f yv = __builtin_nontemporal_load(&y4[q]);

        const int e   = q << 2;            // flat element index of component 0
        const int col = e & 1023;          // multiple of 4; chunk never spans rows
        const int row = (e >> 10) & 1023;
        const float wi = wgt(row);

        const float d0 = yv[0] - xv[0];
        const float d1 = yv[1] - xv[1];
        const float d2 = yv[2] - xv[2];
        const float d3 = yv[3] - xv[3];

        acc0 = fmaf(wi * wgt(col + 0), d0 * d0, acc0);
        acc1 = fmaf(wi * wgt(col + 1), d1 * d1, acc1);
        acc0 = fmaf(wi * wgt(col + 2), d2 * d2, acc0);
        acc1 = fmaf(wi * wgt(col + 3), d3 * d3, acc1);
    }

    // ---- wave32 reduction on the matrix pipe ----
    // A (16x4 f32): lanes 0-15 rows M=0..15 hold {K0,K1}, lanes 16-31 hold
    // {K2,K3}. With A vgpr0=acc0, vgpr1=acc1 and B = ones(4x16):
    //   D[m][n] = acc0[m] + acc1[m] + acc0[m+16] + acc1[m+16]  (indep. of n)
    // Lane k (k<16) D-vgprs 0..7 = rows 0..7; lane k+16 = rows 8..15.
    v2f amat;  amat[0] = acc0;  amat[1] = acc1;
    v2f bones; bones[0] = 1.0f; bones[1] = 1.0f;
    v8f dacc = {};
    dacc = __builtin_amdgcn_wmma_f32_16x16x4_f32(
        /*neg_a=*/false, amat, /*neg_b=*/false, bones,
        /*c_mod=*/(short)0, dacc, /*reuse_a=*/false, /*reuse_b=*/false);

    float t = ((dacc[0] + dacc[1]) + (dacc[2] + dacc[3])) +
              ((dacc[4] + dacc[5]) + (dacc[6] + dacc[7]));
    // rows 0..7 live in lanes 0..15, rows 8..15 in lanes 16..31 -> one xor-16
    t += __shfl_xor(t, 16, 32);   // every lane now holds the full wave sum

    // ---- cross-wave (8 wave32 per block) via LDS, one atomic per block ----
    __shared__ float smem[8];
    const int lane = threadIdx.x & 31;
    const int wv   = threadIdx.x >> 5;
    if (lane == 0) smem[wv] = t;
    __syncthreads();
    if (threadIdx.x == 0) {
        float s = 0.0f;
#pragma unroll
        for (int i = 0; i < 8; ++i) s += smem[i];
        atomicAdd(accum, s);       // global_atomic_add_f32, SCOPE_DEV
    }
}

__global__ void finalize_kernel(const float* __restrict__ accum,
                                float* __restrict__ out) {
    if (threadIdx.x == 0 && blockIdx.x == 0) {
        // mean over 4 * 1022*1022 patches * 9 elements = 37,601,424 terms
        out[0] = accum[0] * (1.0f / 37601424.0f);
    }
}

extern "C" void kernel_launch(void* const* d_in, const int* in_sizes, int n_in,
                              void* d_out, int out_size, void* d_ws, size_t ws_size,
                              hipStream_t stream) {
    (void)in_sizes; (void)n_in; (void)out_size; (void)ws_size;
    const float* x = (const float*)d_in[0];
    const float* y = (const float*)d_in[1];
    float* out = (float*)d_out;
    float* acc = (float*)d_ws;   // one fp32 accumulator in scratch

    zero_accum_kernel<<<1, 1, 0, stream>>>(acc);

    // 1024 blocks x 256 threads (8 wave32 each) -> 262,144 threads,
    // exactly 4 float4 chunks per thread over the 1,048,576 chunks.
    mse_patch_wsum_kernel<<<1024, 256, 0, stream>>>(x, y, acc);

    finalize_kernel<<<1, 1, 0, stream>>>(acc, out);
}